// NLM_48309792145996
// MI455X (gfx1250) — compile-verified
//
#include <hip/hip_runtime.h>
#include <hip/hip_bf16.h>

typedef __attribute__((ext_vector_type(2))) float v2f;
typedef __attribute__((ext_vector_type(8))) float v8f;

#define B_  4
#define C_  128
#define IC_ 64
#define H_  64
#define W_  64
#define HW_ 4096

__device__ __forceinline__ v8f wmma4(v2f a, v2f b, v8f c) {
  // D = A(16x4 f32) * B(4x16 f32) + C(16x16 f32), exact f32 matrix op
  return __builtin_amdgcn_wmma_f32_16x16x4_f32(false, a, false, b, (short)0, c,
                                               false, false);
}
__device__ __forceinline__ v2f ld2(const float* p) {
  return *reinterpret_cast<const v2f*>(p);
}
__device__ __forceinline__ void st2(float* p, v2f v) {
  *reinterpret_cast<v2f*>(p) = v;
}

// ---------------------------------------------------------------------------
// Prep: fold w5/w3 (centered) + 2*identity into one 7x7 kernel, repacked
// pair-interleaved along input channel: wpackI[khw][i/2][o][i&1].
// Every WMMA A-fragment is then a single aligned global_load_b64.
// ---------------------------------------------------------------------------
__global__ void k_prep_wpack(const float* __restrict__ w7, const float* __restrict__ w5,
                             const float* __restrict__ w3, float* __restrict__ wpack) {
  int idx = blockIdx.x * blockDim.x + threadIdx.x;
  const int total = 49 * C_ * C_;
  if (idx >= total) return;
  int khw = idx >> 14;
  int rem = idx & 16383;
  int p   = rem >> 8;          // i/2
  int o   = (rem & 255) >> 1;
  int par = rem & 1;
  int i   = 2 * p + par;
  int kh = khw / 7, kw = khw % 7;
  float v = w7[((o * C_ + i) * 7 + kh) * 7 + kw];
  if (kh >= 1 && kh <= 5 && kw >= 1 && kw <= 5)
    v += w5[((o * C_ + i) * 5 + (kh - 1)) * 5 + (kw - 1)];
  if (kh >= 2 && kh <= 4 && kw >= 2 && kw <= 4)
    v += w3[((o * C_ + i) * 3 + (kh - 2)) * 3 + (kw - 2)];
  if (kh == 3 && kw == 3 && o == i) v += 2.0f;   // both +x residuals folded in
  wpack[idx] = v;
}

// Stacked theta/phi/g 1x1 weights, interleaved: w3tI[c/2][m(0..191)][c&1]
__global__ void k_prep_w3t(const float* __restrict__ tw, const float* __restrict__ pw,
                           const float* __restrict__ gw, float* __restrict__ w3t) {
  int idx = blockIdx.x * blockDim.x + threadIdx.x;
  if (idx >= C_ * 192) return;
  int p   = idx / 384;
  int rem = idx % 384;
  int m   = rem >> 1;
  int c   = 2 * p + (rem & 1);
  float v;
  if (m < 64)       v = tw[m * C_ + c];
  else if (m < 128) v = pw[(m - 64) * C_ + c];
  else              v = gw[(m - 128) * C_ + c];
  w3t[idx] = v;
}

// owtI[c/2][o][c&1] = ow[o][c]
__global__ void k_prep_owt(const float* __restrict__ ow, float* __restrict__ owt) {
  int idx = blockIdx.x * blockDim.x + threadIdx.x;
  if (idx >= IC_ * C_) return;
  int p = idx >> 8;
  int o = (idx & 255) >> 1;
  int c = 2 * p + (idx & 1);
  owt[idx] = ow[o * IC_ + c];
}

// ---------------------------------------------------------------------------
// Fused 7x7 conv (implicit GEMM). One WG per (b,h): M=128 out-ch, N=64 width.
// 4 waves x 2 M-tiles each (B fragments reused by 2 WMMAs). Input rows staged
// in LDS pair-interleaved: lds[c/2][col(0..69)][c&1] -> B frag = ds_load_b64.
// ---------------------------------------------------------------------------
__global__ void __launch_bounds__(128) k_conv(const float* __restrict__ x,
                                              const float* __restrict__ wpack,
                                              const float* __restrict__ b7,
                                              const float* __restrict__ b5,
                                              const float* __restrict__ b3,
                                              float* __restrict__ convS) {
  __shared__ __align__(16) float lds[64 * 140];
  const int bb   = blockIdx.x >> 6;
  const int h    = blockIdx.x & 63;
  const int tid  = threadIdx.x;
  const int wave = tid >> 5;
  const int lane = tid & 31;
  const int half = lane >> 4;
  const int lm   = lane & 15;
  const int m0a  = wave * 16;
  const int m0b  = 64 + wave * 16;

  v8f zero = {};
  v8f acc[2][4];
  for (int g = 0; g < 2; ++g)
    for (int nt = 0; nt < 4; ++nt) acc[g][nt] = zero;

  for (int kh = 0; kh < 7; ++kh) {
    int hy = h + kh - 3;
    if (hy < 0 || hy >= H_) continue;           // uniform across the block
    __syncthreads();
    // stage padded rows as channel-pairs: one b64 LDS store per (pair,col)
    for (int e = tid; e < 64 * 70; e += 128) {
      int p = e / 70;
      int col = e - p * 70;
      int wsrc = col - 3;
      v2f v = {};
      if (wsrc >= 0 && wsrc < W_) {
        const float* xp = x + ((bb * C_ + 2 * p) * H_ + hy) * W_ + wsrc;
        v.x = xp[0];
        v.y = xp[HW_];          // channel 2p+1, same (hy,wsrc)
      }
      st2(&lds[p * 140 + col * 2], v);
    }
    __syncthreads();
    for (int i0 = 0; i0 < C_; i0 += 4) {
      const int pk = (i0 >> 1) + half;          // pair index for k0=i0+2*half
      for (int kw = 0; kw < 7; ++kw) {
        const float* wp = wpack + (kh * 7 + kw) * (C_ * C_) + pk * 256;
        v2f a0 = ld2(wp + (m0a + lm) * 2);
        v2f a1 = ld2(wp + (m0b + lm) * 2);
        #pragma unroll
        for (int nt = 0; nt < 4; ++nt) {
          v2f bf = ld2(&lds[pk * 140 + (nt * 16 + lm + kw) * 2]);
          acc[0][nt] = wmma4(a0, bf, acc[0][nt]);
          acc[1][nt] = wmma4(a1, bf, acc[1][nt]);
        }
      }
    }
  }
  #pragma unroll
  for (int g = 0; g < 2; ++g)
    #pragma unroll
    for (int nt = 0; nt < 4; ++nt)
      #pragma unroll
      for (int r = 0; r < 8; ++r) {
        int row = (g ? m0b : m0a) + r + 8 * half;
        float bias = b7[row] + b5[row] + b3[row];
        convS[((bb * C_ + row) * H_ + h) * W_ + nt * 16 + lm] = acc[g][nt][r] + bias;
      }
}

// ---------------------------------------------------------------------------
// Fused theta/phi/g 1x1 conv: M=192, K=128, N=64 per WG. x block staged in
// LDS pair-interleaved. Outputs stored channel-pair interleaved:
//   thetaI/phiI : [b][d/2][n][d&1]      (B/A operands of S^T GEMM)
//   gTI         : [b][key/2][c][key&1]  (A-operand of attention-out GEMM)
// ---------------------------------------------------------------------------
__global__ void __launch_bounds__(128) k_qkv(const float* __restrict__ x,
                                             const float* __restrict__ w3t,
                                             const float* __restrict__ tb,
                                             const float* __restrict__ pb,
                                             const float* __restrict__ gb,
                                             float* __restrict__ thetaI,
                                             float* __restrict__ phiI,
                                             float* __restrict__ gTI) {
  __shared__ __align__(16) float xI[64 * 128];   // [c/2][col][c&1], 32KB
  const int nglob = blockIdx.x * 64;
  const int bb = nglob >> 12;
  const int hw = nglob & 4095;
  const int tid  = threadIdx.x;
  const int wave = tid >> 5;
  const int lane = tid & 31;
  const int half = lane >> 4;
  const int lm   = lane & 15;

  const float* xb = x + bb * (C_ * HW_) + hw;
  for (int e = tid; e < 64 * 64; e += 128) {
    int p = e >> 6;
    int col = e & 63;
    v2f v;
    v.x = xb[(2 * p) * HW_ + col];
    v.y = xb[(2 * p + 1) * HW_ + col];
    st2(&xI[p * 128 + col * 2], v);
  }
  __syncthreads();

  v8f zero = {};
  v8f acc[3][4];
  for (int j = 0; j < 3; ++j)
    for (int nt = 0; nt < 4; ++nt) acc[j][nt] = zero;

  for (int k0c = 0; k0c < C_; k0c += 4) {
    const int pk = (k0c >> 1) + half;
    v2f bf[4];
    #pragma unroll
    for (int nt = 0; nt < 4; ++nt)
      bf[nt] = ld2(&xI[pk * 128 + (nt * 16 + lm) * 2]);
    #pragma unroll
    for (int j = 0; j < 3; ++j) {
      int m0 = (wave + 4 * j) * 16;
      v2f a = ld2(&w3t[pk * 384 + (m0 + lm) * 2]);
      #pragma unroll
      for (int nt = 0; nt < 4; ++nt)
        acc[j][nt] = wmma4(a, bf[nt], acc[j][nt]);
    }
  }
  #pragma unroll
  for (int j = 0; j < 3; ++j) {
    int m0 = (wave + 4 * j) * 16;
    #pragma unroll
    for (int nt = 0; nt < 4; ++nt)
      #pragma unroll
      for (int r = 0; r < 8; ++r) {
        int row = m0 + r + 8 * half;
        int n = hw + nt * 16 + lm;
        float v = acc[j][nt][r];
        if (row < 64)
          thetaI[((bb * 32 + (row >> 1)) * HW_ + n) * 2 + (row & 1)] = v + tb[row];
        else if (row < 128) {
          int d = row - 64;
          phiI[((bb * 32 + (d >> 1)) * HW_ + n) * 2 + (d & 1)] = v + pb[d];
        } else {
          int c = row - 128;
          gTI[((bb * 2048 + (n >> 1)) * IC_ + c) * 2 + (n & 1)] = v + gb[c];
        }
      }
  }
}

// ---------------------------------------------------------------------------
// Flash attention: per WG one batch's 64-query block; 4 waves.
// Phase 1 (wave owns 16 queries): S^T(64k x 16q) = phi^T theta, all operand
// fragments single b64 loads from pair-interleaved thetaI/phiI. Online
// softmax per query (lanes l and l^16 share a query column). P^T goes
// through pair-interleaved LDS (C-layout -> B-layout conversion).
// Phase 2 (wave owns 16 channels): out = alpha*out + g * P^T.
// Never materializes the 268MB f matrix.
// ---------------------------------------------------------------------------
__global__ void __launch_bounds__(128) k_attn(const float* __restrict__ thetaI,
                                              const float* __restrict__ phiI,
                                              const float* __restrict__ gTI,
                                              float* __restrict__ attnI) {
  __shared__ __align__(16) float ldsP[32 * 128];   // [key/2][q][key&1]
  __shared__ float ldsA[64];
  const int bb = blockIdx.x >> 6;
  const int q0 = (blockIdx.x & 63) * 64;
  const int tid  = threadIdx.x;
  const int wave = tid >> 5;
  const int lane = tid & 31;
  const int half = lane >> 4;
  const int lm   = lane & 15;
  const int cw   = wave * 16;        // phase-2 channel slice
  const int qw   = q0 + wave * 16;   // phase-1 query slice

  v8f zero = {};
  v8f oacc[4];
  for (int nt = 0; nt < 4; ++nt) oacc[nt] = zero;
  float Mrun = -3.0e38f;
  float Lrun = 0.0f;

  const float* thb = thetaI + bb * (32 * HW_ * 2);
  const float* phb = phiI + bb * (32 * HW_ * 2);
  const float* gtb = gTI + bb * (2048 * IC_ * 2);

  for (int kb = 0; kb < 64; ++kb) {
    const int K0 = kb * 64;
    // ---- phase 1
    v8f sacc[4];
    for (int mt = 0; mt < 4; ++mt) sacc[mt] = zero;
    for (int d0 = 0; d0 < IC_; d0 += 4) {
      const int pd = (d0 >> 1) + half;
      v2f bf = ld2(&thb[(pd * HW_ + qw + lm) * 2]);
      #pragma unroll
      for (int mt = 0; mt < 4; ++mt) {
        v2f a = ld2(&phb[(pd * HW_ + K0 + mt * 16 + lm) * 2]);
        sacc[mt] = wmma4(a, bf, sacc[mt]);
      }
    }
    float bm = -3.0e38f;
    #pragma unroll
    for (int mt = 0; mt < 4; ++mt)
      #pragma unroll
      for (int r = 0; r < 8; ++r) {
        float s = sacc[mt][r] * 0.125f;   // 1/sqrt(IC)
        sacc[mt][r] = s;
        bm = fmaxf(bm, s);
      }
    bm = fmaxf(bm, __shfl_xor(bm, 16, 32));
    float Mnew = fmaxf(Mrun, bm);
    float alpha = __expf(Mrun - Mnew);
    float bs = 0.0f;
    #pragma unroll
    for (int mt = 0; mt < 4; ++mt)
      #pragma unroll
      for (int r = 0; r < 8; ++r) {
        float p = __expf(sacc[mt][r] - Mnew);
        sacc[mt][r] = p;
        bs += p;
      }
    bs += __shfl_xor(bs, 16, 32);
    Lrun = Lrun * alpha + bs;
    Mrun = Mnew;

    __syncthreads();                      // prior phase-2 done with LDS
    #pragma unroll
    for (int mt = 0; mt < 4; ++mt)
      #pragma unroll
      for (int r = 0; r < 8; ++r) {
        int key = mt * 16 + r + 8 * half;
        ldsP[(key >> 1) * 128 + (wave * 16 + lm) * 2 + (key & 1)] = sacc[mt][r];
      }
    if (lane < 16) ldsA[wave * 16 + lane] = alpha;
    __syncthreads();

    // ---- phase 2
    #pragma unroll
    for (int nt = 0; nt < 4; ++nt) {
      float al = ldsA[nt * 16 + lm];
      #pragma unroll
      for (int r = 0; r < 8; ++r) oacc[nt][r] *= al;
    }
    for (int k0 = 0; k0 < 64; k0 += 4) {
      const int pk = (k0 >> 1) + half;
      v2f a = ld2(&gtb[(((K0 >> 1) + pk) * IC_ + cw + lm) * 2]);
      #pragma unroll
      for (int nt = 0; nt < 4; ++nt) {
        v2f bf = ld2(&ldsP[pk * 128 + (nt * 16 + lm) * 2]);
        oacc[nt] = wmma4(a, bf, oacc[nt]);
      }
    }
  }
  __syncthreads();
  if (lane < 16) ldsA[wave * 16 + lane] = Lrun;
  __syncthreads();
  #pragma unroll
  for (int nt = 0; nt < 4; ++nt) {
    float rinv = 1.0f / ldsA[nt * 16 + lm];
    #pragma unroll
    for (int r = 0; r < 8; ++r) {
      int row = cw + r + 8 * half;
      int n = q0 + nt * 16 + lm;
      attnI[((bb * 32 + (row >> 1)) * HW_ + n) * 2 + (row & 1)] = oacc[nt][r] * rinv;
    }
  }
}

// ---------------------------------------------------------------------------
// Final projection ow(128x64)*attn + ob + convS, relu. 4 waves x 2 M-tiles.
// attnI is channel-pair interleaved -> B frag = single global_load_b64.
// ---------------------------------------------------------------------------
__global__ void __launch_bounds__(128) k_final(const float* __restrict__ attnI,
                                               const float* __restrict__ owt,
                                               const float* __restrict__ ob,
                                               const float* __restrict__ convS,
                                               float* __restrict__ out) {
  const int nglob = blockIdx.x * 64;
  const int bb = nglob >> 12;
  const int hw = nglob & 4095;
  const int tid  = threadIdx.x;
  const int wave = tid >> 5;
  const int lane = tid & 31;
  const int half = lane >> 4;
  const int lm   = lane & 15;
  const int m0a  = wave * 16;
  const int m0b  = 64 + wave * 16;

  v8f zero = {};
  v8f acc[2][4];
  for (int g = 0; g < 2; ++g)
    for (int nt = 0; nt < 4; ++nt) acc[g][nt] = zero;

  const float* ab = attnI + bb * (32 * HW_ * 2);
  for (int k0c = 0; k0c < IC_; k0c += 4) {
    const int pk = (k0c >> 1) + half;
    v2f a0 = ld2(&owt[pk * 256 + (m0a + lm) * 2]);
    v2f a1 = ld2(&owt[pk * 256 + (m0b + lm) * 2]);
    #pragma unroll
    for (int nt = 0; nt < 4; ++nt) {
      v2f bf = ld2(&ab[(pk * HW_ + hw + nt * 16 + lm) * 2]);
      acc[0][nt] = wmma4(a0, bf, acc[0][nt]);
      acc[1][nt] = wmma4(a1, bf, acc[1][nt]);
    }
  }
  #pragma unroll
  for (int g = 0; g < 2; ++g)
    #pragma unroll
    for (int nt = 0; nt < 4; ++nt)
      #pragma unroll
      for (int r = 0; r < 8; ++r) {
        int row = (g ? m0b : m0a) + r + 8 * half;
        int idx = (bb * C_ + row) * HW_ + hw + nt * 16 + lm;
        float v = acc[g][nt][r] + ob[row] + convS[idx];
        out[idx] = fmaxf(v, 0.0f);
      }
}

// ---------------------------------------------------------------------------
extern "C" void kernel_launch(void* const* d_in, const int* in_sizes, int n_in,
                              void* d_out, int out_size, void* d_ws, size_t ws_size,
                              hipStream_t stream) {
  const float* x  = (const float*)d_in[0];
  const float* w7 = (const float*)d_in[1];
  const float* b7 = (const float*)d_in[2];
  const float* w5 = (const float*)d_in[3];
  const float* b5 = (const float*)d_in[4];
  const float* w3 = (const float*)d_in[5];
  const float* b3 = (const float*)d_in[6];
  const float* gw = (const float*)d_in[7];
  const float* gb = (const float*)d_in[8];
  const float* tw = (const float*)d_in[9];
  const float* tb = (const float*)d_in[10];
  const float* pw = (const float*)d_in[11];
  const float* pb = (const float*)d_in[12];
  const float* ow = (const float*)d_in[13];
  const float* ob = (const float*)d_in[14];

  float* ws     = (float*)d_ws;
  float* wpack  = ws;                 // 49*128*128 = 802816
  float* w3t    = wpack + 802816;     // 128*192   = 24576
  float* owt    = w3t + 24576;        // 64*128    = 8192
  float* thetaI = owt + 8192;         // 4*64*4096 = 1048576
  float* phiI   = thetaI + 1048576;
  float* gTI    = phiI + 1048576;
  float* attnI  = gTI + 1048576;
  float* convS  = attnI + 1048576;    // 4*128*4096 = 2097152  (total ~28.5 MB)
  float* outf   = (float*)d_out;

  k_prep_wpack<<<(49 * C_ * C_ + 255) / 256, 256, 0, stream>>>(w7, w5, w3, wpack);
  k_prep_w3t<<<(C_ * 192 + 255) / 256, 256, 0, stream>>>(tw, pw, gw, w3t);
  k_prep_owt<<<(IC_ * C_ + 255) / 256, 256, 0, stream>>>(ow, owt);
  k_qkv<<<256, 128, 0, stream>>>(x, w3t, tb, pb, gb, thetaI, phiI, gTI);
  k_conv<<<256, 128, 0, stream>>>(x, wpack, b7, b5, b3, convS);
  k_attn<<<256, 128, 0, stream>>>(thetaI, phiI, gTI, attnI);
  k_final<<<256, 128, 0, stream>>>(attnI, owt, ob, convS, outf);
}